// PointcloudGrouping_78993038508353
// MI455X (gfx1250) — compile-verified
//
#include <hip/hip_runtime.h>

#define NPTS  16384
#define NSEED 512
#define KNN   32
#define CHUNK 2048
#define NB    8

typedef __attribute__((ext_vector_type(2))) float v2f;
typedef __attribute__((ext_vector_type(8))) float v8f;

// ---------------------------------------------------------------------------
// Kernel 1: farthest point sampling. One block per batch, 256 threads (8 waves).
// Sequential in K by construction; each step is an L2-resident 16384-pt scan.
// ---------------------------------------------------------------------------
__global__ __launch_bounds__(256) void fps_kernel(const float4* __restrict__ pts,
                                                  float* __restrict__ seed_xyz)
{
    __shared__ float s_val[256];
    __shared__ int   s_idx[256];
    __shared__ float s_last[3];

    const int b   = blockIdx.x;
    const int tid = threadIdx.x;
    const float4* p = pts + (size_t)b * NPTS;

    float mind[NPTS / 256];
#pragma unroll
    for (int i = 0; i < NPTS / 256; ++i) mind[i] = __builtin_inff();

    if (tid == 0) {
        float4 q = p[0];
        s_last[0] = q.x; s_last[1] = q.y; s_last[2] = q.z;
    }
    __syncthreads();

    for (int k = 0; k < NSEED; ++k) {
        if (tid == 0) {   // record current seed (pre-update, matches lax.scan order)
            size_t o = ((size_t)b * NSEED + k) * 3;
            seed_xyz[o + 0] = s_last[0];
            seed_xyz[o + 1] = s_last[1];
            seed_xyz[o + 2] = s_last[2];
        }
        const float lx = s_last[0], ly = s_last[1], lz = s_last[2];
        float bv = -1.0f; int bj = 0x7fffffff;
        for (int i = 0; i < NPTS / 256; ++i) {
            int j = tid + (i << 8);
            float4 q = p[j];
            float dx = q.x - lx, dy = q.y - ly, dz = q.z - lz;
            float d = dx * dx + dy * dy + dz * dz;
            float m = mind[i];
            m = d < m ? d : m;
            mind[i] = m;
            bool t = m > bv;                  // strict '>' keeps lowest j on ties
            bv = t ? m : bv;
            bj = t ? j : bj;
        }
        s_val[tid] = bv; s_idx[tid] = bj;
        __syncthreads();
        for (int off = 128; off > 0; off >>= 1) {   // argmax, tie -> lower index
            if (tid < off) {
                float v2 = s_val[tid + off]; int j2 = s_idx[tid + off];
                bool t = (v2 > s_val[tid]) | ((v2 == s_val[tid]) & (j2 < s_idx[tid]));
                if (t) { s_val[tid] = v2; s_idx[tid] = j2; }
            }
            __syncthreads();
        }
        if (tid == 0) {
            float4 q = p[s_idx[0]];
            s_last[0] = q.x; s_last[1] = q.y; s_last[2] = q.z;
        }
        __syncthreads();
    }
}

// ---------------------------------------------------------------------------
// Kernel 2: KNN (top-32 smallest d2) + grouping + centers.
// One block per (batch, 16-seed tile). Distances via V_WMMA_F32_16X16X4_F32:
//   d2' = ||p||^2 - 2 s.p   (per-seed constant ||s||^2 dropped; per-seed
//   ranking unchanged).  A row s = [sx,sy,sz,1],  B col p = [-2px,-2py,-2pz,|p|^2].
// Each seed group = one half-wave (16 lanes): the top-K merge runs entirely
// intra-wave (shuffles + same-wave LDS ordering), no per-round barriers.
// ---------------------------------------------------------------------------
__global__ __launch_bounds__(256) void knn_group_kernel(const float4* __restrict__ pts,
                                                        const float* __restrict__ seed_xyz,
                                                        float* __restrict__ groups_out,
                                                        float* __restrict__ centers_out)
{
    extern __shared__ float s_d2[];            // [16][CHUNK]  (dynamic, 128 KB)
    __shared__ float s_best_val[16][KNN];
    __shared__ int   s_best_idx[16][KNN];
    __shared__ float s_new_val[16][KNN];
    __shared__ int   s_new_idx[16][KNN];
    __shared__ float s_seed[16][3];

    const int b    = blockIdx.x >> 5;          // 8 batches x 32 tiles
    const int tile = blockIdx.x & 31;
    const int tid  = threadIdx.x;
    const float4* p = pts + (size_t)b * NPTS;

    const int lane = tid & 31;
    const int wave = __builtin_amdgcn_readfirstlane(tid >> 5);  // uniform SGPR
    const int half = lane >> 4;                // A/B K-half selector (ISA 7.12.2)
    const int l16  = lane & 15;
    const int g    = tid >> 4;                 // seed group (0..15) = half-wave
    const int li   = tid & 15;

    if (tid < 48) {                            // 16 seeds x 3 coords
        int s = tid / 3, c = tid % 3;
        s_seed[s][c] = seed_xyz[((size_t)b * NSEED + tile * 16 + s) * 3 + c];
    }
    for (int c = li; c < KNN; c += 16) {       // per-group init (own arrays only)
        s_best_val[g][c] = __builtin_inff();
        s_best_idx[g][c] = 0x7fffffff;
    }
    __syncthreads();

    v2f afrag;
    {
        float sx = s_seed[l16][0], sy = s_seed[l16][1], sz = s_seed[l16][2];
        afrag[0] = half ? sz   : sx;           // lo lanes: K=0,1  hi lanes: K=2,3
        afrag[1] = half ? 1.0f : sy;
    }

    for (int cb = 0; cb < NPTS; cb += CHUNK) {
        // ---- WMMA: fill 16 x CHUNK d2 tile; wave w handles tiles w, w+8, ...
#pragma unroll 2
        for (int t = wave; t < CHUNK / 16; t += 8) {
            float4 q = p[cb + t * 16 + l16];
            float n2 = q.x * q.x + q.y * q.y + q.z * q.z;
            v2f bfrag;
            bfrag[0] = half ? (-2.0f * q.z) : (-2.0f * q.x);
            bfrag[1] = half ? n2            : (-2.0f * q.y);
            v8f acc = {};
            acc = __builtin_amdgcn_wmma_f32_16x16x4_f32(false, afrag, false, bfrag,
                                                        (short)0, acc, false, false);
#pragma unroll
            for (int v = 0; v < 8; ++v) {      // C layout: VGPR v -> M = v + 8*half
                int m = v + 8 * half;
                s_d2[m * CHUNK + t * 16 + l16] = acc[v];
            }
        }
        __syncthreads();                       // all rows written before scans

        // ---- merge chunk into running top-K (iterative argmin == top_k order)
        const float4* row4 = (const float4*)(s_d2 + g * CHUNK);
        for (int r = 0; r < KNN; ++r) {
            float bv = __builtin_inff(); int bi = 0x7fffffff; int bs = -1;
            // chunk scan: thread li owns float4 slots li, li+16, ... (ds_load_b128,
            // conflict-free: 16 consecutive float4s span all 64 banks)
            for (int i = 0; i < CHUNK / 64; ++i) {
                int f4 = i * 16 + li;
                float4 vq = row4[f4];
                int gi0 = cb + f4 * 4;
                {   bool t = (vq.x < bv) | ((vq.x == bv) & (gi0 < bi));
                    bv = t ? vq.x : bv; bi = t ? gi0 : bi; bs = t ? f4 * 4 + 0 : bs; }
                {   int gi = gi0 + 1;
                    bool t = (vq.y < bv) | ((vq.y == bv) & (gi < bi));
                    bv = t ? vq.y : bv; bi = t ? gi : bi; bs = t ? f4 * 4 + 1 : bs; }
                {   int gi = gi0 + 2;
                    bool t = (vq.z < bv) | ((vq.z == bv) & (gi < bi));
                    bv = t ? vq.z : bv; bi = t ? gi : bi; bs = t ? f4 * 4 + 2 : bs; }
                {   int gi = gi0 + 3;
                    bool t = (vq.w < bv) | ((vq.w == bv) & (gi < bi));
                    bv = t ? vq.w : bv; bi = t ? gi : bi; bs = t ? f4 * 4 + 3 : bs; }
            }
            // carry scan (previous chunks' survivors; their indices are smaller)
            for (int c = li; c < KNN; c += 16) {
                float v = s_best_val[g][c]; int gi = s_best_idx[g][c];
                bool t = (v < bv) | ((v == bv) & (gi < bi));
                bv = t ? v : bv; bi = t ? gi : bi; bs = t ? CHUNK + c : bs;
            }
            // half-wave reduction: winner broadcast to all 16 lanes of the group
#pragma unroll
            for (int off = 8; off > 0; off >>= 1) {
                float ov = __shfl_xor(bv, off, 16);
                int   oi = __shfl_xor(bi, off, 16);
                int   os = __shfl_xor(bs, off, 16);
                bool t = (ov < bv) | ((ov == bv) & (oi < bi));
                bv = t ? ov : bv; bi = t ? oi : bi; bs = t ? os : bs;
            }
            if (li == 0) {
                s_new_val[g][r] = bv; s_new_idx[g][r] = bi;
                if (bs >= CHUNK) s_best_val[g][bs - CHUNK] = __builtin_inff();
                else             s_d2[g * CHUNK + bs]      = __builtin_inff();
            }
            // same-wave DS ordering: store above is seen by this group's loads
        }
        for (int c = li; c < KNN; c += 16) {
            s_best_val[g][c] = s_new_val[g][c];
            s_best_idx[g][c] = s_new_idx[g][c];
        }
        __syncthreads();                       // before next fill overwrites s_d2
    }

    // ---- centers (mean of group xyz) via half-wave shuffle reduction
    float sx = 0.f, sy = 0.f, sz = 0.f;
    for (int j = li; j < KNN; j += 16) {
        float4 q = p[s_best_idx[g][j]];
        sx += q.x; sy += q.y; sz += q.z;
    }
#pragma unroll
    for (int off = 8; off > 0; off >>= 1) {
        sx += __shfl_xor(sx, off, 16);
        sy += __shfl_xor(sy, off, 16);
        sz += __shfl_xor(sz, off, 16);
    }
    const float cx = sx / (float)KNN;
    const float cy = sy / (float)KNN;
    const float cz = sz / (float)KNN;

    const size_t sg = (size_t)b * NSEED + tile * 16 + g;
    for (int j = li; j < KNN; j += 16) {
        float4 q = p[s_best_idx[g][j]];
        size_t o = (sg * KNN + j) * 4;
        groups_out[o + 0] = q.x - cx;
        groups_out[o + 1] = q.y - cy;
        groups_out[o + 2] = q.z - cz;
        groups_out[o + 3] = q.w;
    }
    if (li == 0) {
        centers_out[sg * 3 + 0] = cx;
        centers_out[sg * 3 + 1] = cy;
        centers_out[sg * 3 + 2] = cz;
    }
}

// ---------------------------------------------------------------------------
extern "C" void kernel_launch(void* const* d_in, const int* in_sizes, int n_in,
                              void* d_out, int out_size, void* d_ws, size_t ws_size,
                              hipStream_t stream) {
    (void)in_sizes; (void)n_in; (void)out_size; (void)ws_size;
    const float4* pts = (const float4*)d_in[0];          // (8, 16384, 4) f32
    float* seed_xyz   = (float*)d_ws;                    // (8, 512, 3) f32 scratch
    float* groups_out = (float*)d_out;                   // (8, 512, 32, 4) f32
    float* centers_out = groups_out + (size_t)NB * NSEED * KNN * 4;  // (8, 512, 3)

    fps_kernel<<<NB, 256, 0, stream>>>(pts, seed_xyz);

    size_t lds = (size_t)16 * CHUNK * sizeof(float);     // 128 KB dynamic LDS
    knn_group_kernel<<<NB * 32, 256, lds, stream>>>(pts, seed_xyz, groups_out, centers_out);
}